// DecomposeMambaSSM_47278999994781
// MI455X (gfx1250) — compile-verified
//
#include <hip/hip_runtime.h>
#include <math.h>

typedef float v2f __attribute__((ext_vector_type(2)));
typedef float v8f __attribute__((ext_vector_type(8)));

#define BDIM   4
#define LDIM   4096
#define BL     (BDIM*LDIM)
#define CIN    128
#define DMODEL 64
#define DINNER 128

__device__ __forceinline__ v8f wmma4(v2f a, v2f b, v8f c) {
  // V_WMMA_F32_16X16X4_F32 : (neg_a, A, neg_b, B, c_mod, C, reuse_a, reuse_b)
  return __builtin_amdgcn_wmma_f32_16x16x4_f32(false, a, false, b, (short)0, c,
                                               false, false);
}

__global__ void k_zero(float* __restrict__ p, int n) {
  int i = blockIdx.x * blockDim.x + threadIdx.x;
  if (i < n) p[i] = 0.0f;
}

// -------- embedding conv (wrap pad, k=3): h[b,l,d] = sum_{c,k} x[b,(l-1+k)%L,c]*W[d,c,k]
// im2col GEMM: M=BL, K=384 (kk=c*3+k), N=64. grid = 1024*4 tiles, 32 thr/blk.
__global__ void k_emb_conv(const float* __restrict__ x, const float* __restrict__ w,
                           float* __restrict__ h) {
  int tile = blockIdx.x;
  int mt = tile >> 2, nt = tile & 3;
  int lane = threadIdx.x & 31;
  int half = lane >> 4, lm = lane & 15;
  int r = mt * 16 + lm;
  int b = r >> 12, l = r & 4095;
  int col = nt * 16 + lm;
  const float* wrow = w + (size_t)col * 384;
  v8f acc = {};
  for (int k = 0; k < 384; k += 4) {
    int kk = k + 2 * half;
    int c0 = kk / 3, q0 = kk - c0 * 3;
    int p0 = l - 1 + q0; p0 += (p0 < 0) ? LDIM : 0; p0 -= (p0 >= LDIM) ? LDIM : 0;
    int kk1 = kk + 1;
    int c1 = kk1 / 3, q1 = kk1 - c1 * 3;
    int p1 = l - 1 + q1; p1 += (p1 < 0) ? LDIM : 0; p1 -= (p1 >= LDIM) ? LDIM : 0;
    v2f av, bv;
    av.x = x[((size_t)(b * LDIM + p0)) * CIN + c0];
    av.y = x[((size_t)(b * LDIM + p1)) * CIN + c1];
    bv.x = wrow[kk];
    bv.y = wrow[kk1];
    acc = wmma4(av, bv, acc);
  }
#pragma unroll
  for (int j = 0; j < 8; ++j) {
    int m = mt * 16 + half * 8 + j;
    h[(size_t)m * DMODEL + col] = acc[j];
  }
}

// -------- rmsnorm: one wave per 64-wide row, 8 rows per 256-thread block
__global__ void k_rmsnorm(const float* __restrict__ h, const float* __restrict__ nw,
                          float* __restrict__ xn) {
  int wid = threadIdx.x >> 5, lane = threadIdx.x & 31;
  int row = blockIdx.x * 8 + wid;
  const float* hr = h + (size_t)row * 64;
  float x0 = hr[lane], x1 = hr[lane + 32];
  float ss = x0 * x0 + x1 * x1;
  for (int off = 16; off >= 1; off >>= 1) ss += __shfl_xor(ss, off, 32);
  float sc = rsqrtf(ss * (1.0f / 64.0f) + 1e-5f);
  float* xr = xn + (size_t)row * 64;
  xr[lane]      = x0 * sc * nw[lane];
  xr[lane + 32] = x1 * sc * nw[lane + 32];
}

// -------- generic D(M x ldd) = A(M x K) * W(Nreal x K)^T  (WMMA f32 16x16x4)
__global__ void k_wmma_gemm_nt(const float* __restrict__ A, int lda,
                               const float* __restrict__ W, int ldw,
                               float* __restrict__ D, int ldd,
                               int K, int Nreal, int ntiles) {
  int tile = blockIdx.x;
  int mt = tile / ntiles, nt = tile % ntiles;
  int lane = threadIdx.x & 31;
  int half = lane >> 4, lm = lane & 15;
  const float* arow = A + (size_t)(mt * 16 + lm) * lda;
  int col = nt * 16 + lm;
  bool wv = col < Nreal;
  const float* wrow = W + (size_t)(wv ? col : 0) * ldw;
  v8f acc = {};
  for (int k = 0; k < K; k += 4) {
    int kk = k + 2 * half;
    v2f av, bv;
    av.x = arow[kk];
    av.y = arow[kk + 1];
    bv.x = wv ? wrow[kk] : 0.0f;
    bv.y = wv ? wrow[kk + 1] : 0.0f;
    acc = wmma4(av, bv, acc);
  }
#pragma unroll
  for (int j = 0; j < 8; ++j) {
    int m = mt * 16 + half * 8 + j;
    D[(size_t)m * ldd + col] = acc[j];
  }
}

// -------- causal depthwise conv (k=4) + bias + SiLU; input xi = xz[:, 0:128]
__global__ void k_dwconv_silu(const float* __restrict__ xz, const float* __restrict__ cw,
                              const float* __restrict__ cb, float* __restrict__ xc) {
  int i = blockIdx.x * blockDim.x + threadIdx.x;
  if (i >= BL * DINNER) return;
  int r = i >> 7, d = i & 127;
  int b = r >> 12, l = r & 4095;
  float acc = cb[d];
#pragma unroll
  for (int k = 0; k < 4; ++k) {
    int ll = l - 3 + k;
    if (ll >= 0) acc += xz[((size_t)(b * LDIM + ll)) * 256 + d] * cw[d * 4 + k];
  }
  xc[(size_t)r * 128 + d] = acc / (1.0f + __expf(-acc));
}

// -------- delta = softplus(dlt @ dtw^T + dtb): K=4 -> single WMMA step
__global__ void k_delta(const float* __restrict__ xbl, const float* __restrict__ dtw,
                        const float* __restrict__ dtb, float* __restrict__ delta) {
  int tile = blockIdx.x;           // 1024 * 8
  int mt = tile >> 3, nt = tile & 7;
  int lane = threadIdx.x & 31;
  int half = lane >> 4, lm = lane & 15;
  const float* arow = xbl + (size_t)(mt * 16 + lm) * 48;   // cols 0..3 = dlt
  int col = nt * 16 + lm;
  const float* wrow = dtw + (size_t)col * 4;
  v2f av, bv;
  av.x = arow[2 * half];     av.y = arow[2 * half + 1];
  bv.x = wrow[2 * half];     bv.y = wrow[2 * half + 1];
  v8f acc = {};
  acc = wmma4(av, bv, acc);
  float bias = dtb[col];
#pragma unroll
  for (int j = 0; j < 8; ++j) {
    int m = mt * 16 + half * 8 + j;
    float v = acc[j] + bias;
    float sp = (v > 20.0f) ? v : log1pf(__expf(v));
    delta[(size_t)m * 128 + col] = sp;
  }
}

// -------- selective scan: lane = state n (16), wave halves = 2 channels; fused gate
__global__ void k_scan(const float* __restrict__ xc, const float* __restrict__ delta,
                       const float* __restrict__ xbl, const float* __restrict__ xz,
                       const float* __restrict__ alog, const float* __restrict__ dpar,
                       float* __restrict__ ys) {
  int lane = threadIdx.x & 31;
  int half = lane >> 4, n = lane & 15;
  int cid = blockIdx.x * 2 + half;   // (b,d) channel id, 512 total
  int b = cid >> 7, d = cid & 127;
  float a  = -__expf(alog[d * 16 + n]);
  float Dp = dpar[d];
  float h = 0.0f;
  size_t base = (size_t)b * LDIM;
  for (int l = 0; l < LDIM; ++l) {
    size_t r = base + l;
    float dl = delta[r * 128 + d];
    float u  = xc[r * 128 + d];
    float Bv = xbl[r * 48 + 4 + n];     // B state
    float Cv = xbl[r * 48 + 20 + n];    // C state
    h = __expf(dl * a) * h + dl * u * Bv;
    float part = h * Cv;
    part += __shfl_xor(part, 8, 16);
    part += __shfl_xor(part, 4, 16);
    part += __shfl_xor(part, 2, 16);
    part += __shfl_xor(part, 1, 16);
    if (n == 0) {
      float z = xz[r * 256 + 128 + d];
      float sz = z / (1.0f + __expf(-z));
      ys[r * 128 + d] = (part + u * Dp) * sz;
    }
  }
}

// -------- radix-2 LDS FFT (4096 pts) per (b,d) series; accumulate PSD
__global__ void k_fft_psd(const float* __restrict__ yo, float* __restrict__ psd) {
  __shared__ float re[4096];
  __shared__ float im[4096];
  int s = blockIdx.x;            // 256 series
  int b = s >> 6, d = s & 63;
  int t = threadIdx.x;           // 256 threads
  for (int i = t; i < 4096; i += 256) {
    re[i] = yo[((size_t)(b * LDIM + i)) * 64 + d];
    im[i] = 0.0f;
  }
  __syncthreads();
  for (int i = t; i < 4096; i += 256) {
    int j = (int)(__brev((unsigned)i) >> 20);
    if (j > i) {
      float tr = re[i]; re[i] = re[j]; re[j] = tr;
      float ti = im[i]; im[i] = im[j]; im[j] = ti;
    }
  }
  __syncthreads();
  for (int len = 2; len <= 4096; len <<= 1) {
    int hl = len >> 1;
    float ang = -6.283185307179586f / (float)len;
    for (int q = t; q < 2048; q += 256) {
      int g = q / hl, pos = q - g * hl;
      int i0 = g * len + pos, i1 = i0 + hl;
      float sn, cs;
      __sincosf(ang * (float)pos, &sn, &cs);
      float xr = re[i1], xi = im[i1];
      float tr = xr * cs - xi * sn;
      float ti = xr * sn + xi * cs;
      re[i1] = re[i0] - tr; im[i1] = im[i0] - ti;
      re[i0] += tr;         im[i0] += ti;
    }
    __syncthreads();
  }
  for (int f = 1 + t; f < 2048; f += 256)
    atomicAdd(&psd[f], re[f] * re[f] + im[f] * im[f]);
}

// -------- argmax over psd[1..2047] -> period p, front offset
__global__ void k_argmax_period(const float* __restrict__ psd, int* __restrict__ prm) {
  __shared__ float bv[1024];
  __shared__ int   bi[1024];
  int t = threadIdx.x;
  float best = -1.0f; int bidx = 1;
  for (int f = 1 + t; f < 2048; f += 1024) {
    float v = psd[f];
    if (v > best) { best = v; bidx = f; }
  }
  bv[t] = best; bi[t] = bidx;
  __syncthreads();
  for (int sft = 512; sft >= 1; sft >>= 1) {
    if (t < sft) {
      if (bv[t + sft] > bv[t] || (bv[t + sft] == bv[t] && bi[t + sft] < bi[t])) {
        bv[t] = bv[t + sft]; bi[t] = bi[t + sft];
      }
    }
    __syncthreads();
  }
  if (t == 0) {
    int f = bi[0];
    int p = 4096 / f;
    int front = (p - 1) / 2 + (((p & 1) == 0) ? 1 : 0);
    prm[0] = p; prm[1] = front;
  }
}

// -------- cumsum per series (exclusive, cs has L+1 entries)
__global__ void k_cumsum(const float* __restrict__ yo, float* __restrict__ cs) {
  __shared__ float part[1024];
  int s = blockIdx.x;  int b = s >> 6, d = s & 63;
  int t = threadIdx.x;
  int base = t * 4;
  float v0 = yo[((size_t)(b * LDIM + base + 0)) * 64 + d];
  float v1 = yo[((size_t)(b * LDIM + base + 1)) * 64 + d];
  float v2 = yo[((size_t)(b * LDIM + base + 2)) * 64 + d];
  float v3 = yo[((size_t)(b * LDIM + base + 3)) * 64 + d];
  float p0 = v0, p1 = p0 + v1, p2 = p1 + v2, p3 = p2 + v3;
  part[t] = p3;
  __syncthreads();
  for (int off = 1; off < 1024; off <<= 1) {
    float add = (t >= off) ? part[t - off] : 0.0f;
    __syncthreads();
    part[t] += add;
    __syncthreads();
  }
  float excl = part[t] - p3;
  float* co = cs + (size_t)s * (LDIM + 1);
  if (t == 0) co[0] = 0.0f;
  co[base + 1] = excl + p0;
  co[base + 2] = excl + p1;
  co[base + 3] = excl + p2;
  co[base + 4] = excl + p3;
}

// -------- h = yo - trend ; trends += trend
__global__ void k_trend_update(const float* __restrict__ yo, const float* __restrict__ cs,
                               const int* __restrict__ prm, float* __restrict__ h,
                               float* __restrict__ trends) {
  int i = blockIdx.x * blockDim.x + threadIdx.x;
  if (i >= BL * 64) return;
  int r = i >> 6, d = i & 63;
  int b = r >> 12, l = r & 4095;
  int p = prm[0], front = prm[1];
  int idx = l - front;
  idx = idx < 0 ? 0 : idx;
  int mx = LDIM - p;
  idx = idx > mx ? mx : idx;
  const float* co = cs + (size_t)(b * 64 + d) * (LDIM + 1);
  float tr = (co[idx + p] - co[idx]) / (float)p;
  float yv = yo[(size_t)r * 64 + d];
  h[(size_t)r * 64 + d] = yv - tr;
  trends[(size_t)r * 64 + d] += tr;
}

// -------- final: out = conv_edge(trends, tw) + conv_edge(h, sw), fused WMMA im2col
__global__ void k_final_conv(const float* __restrict__ trends, const float* __restrict__ hseas,
                             const float* __restrict__ tw, const float* __restrict__ sw,
                             float* __restrict__ out) {
  int tile = blockIdx.x;     // 1024 * 8
  int mt = tile >> 3, nt = tile & 7;
  int lane = threadIdx.x & 31;
  int half = lane >> 4, lm = lane & 15;
  int r = mt * 16 + lm;
  int b = r >> 12, l = r & 4095;
  int col = nt * 16 + lm;
  const float* twr = tw + (size_t)col * 192;
  const float* swr = sw + (size_t)col * 192;
  v8f acc = {};
  for (int k = 0; k < 192; k += 4) {
    int kk = k + 2 * half;
    int d0 = kk / 3, q0 = kk - d0 * 3;
    int pos0 = l - 1 + q0; pos0 = pos0 < 0 ? 0 : (pos0 > 4095 ? 4095 : pos0);
    int kk1 = kk + 1;
    int d1 = kk1 / 3, q1 = kk1 - d1 * 3;
    int pos1 = l - 1 + q1; pos1 = pos1 < 0 ? 0 : (pos1 > 4095 ? 4095 : pos1);
    size_t o0 = ((size_t)(b * LDIM + pos0)) * 64 + d0;
    size_t o1 = ((size_t)(b * LDIM + pos1)) * 64 + d1;
    v2f av, bv;
    av.x = trends[o0]; av.y = trends[o1];
    bv.x = twr[kk];    bv.y = twr[kk1];
    acc = wmma4(av, bv, acc);
    av.x = hseas[o0];  av.y = hseas[o1];
    bv.x = swr[kk];    bv.y = swr[kk1];
    acc = wmma4(av, bv, acc);
  }
#pragma unroll
  for (int j = 0; j < 8; ++j) {
    int m = mt * 16 + half * 8 + j;
    out[(size_t)m * CIN + col] = acc[j];
  }
}

extern "C" void kernel_launch(void* const* d_in, const int* in_sizes, int n_in,
                              void* d_out, int out_size, void* d_ws, size_t ws_size,
                              hipStream_t stream) {
  (void)in_sizes; (void)n_in; (void)out_size; (void)ws_size;
  const float* x    = (const float*)d_in[0];
  const float* embw = (const float*)d_in[1];
  const float* nw   = (const float*)d_in[2];
  const float* ipw  = (const float*)d_in[3];
  const float* cw   = (const float*)d_in[4];
  const float* cb   = (const float*)d_in[5];
  const float* xpw  = (const float*)d_in[6];
  const float* dtw  = (const float*)d_in[7];
  const float* dtb  = (const float*)d_in[8];
  const float* alog = (const float*)d_in[9];
  const float* dpar = (const float*)d_in[10];
  const float* opw  = (const float*)d_in[11];
  const float* sw   = (const float*)d_in[12];
  const float* tw   = (const float*)d_in[13];
  float* out = (float*)d_out;

  float* ws = (float*)d_ws;
  size_t off = 0;
  float* f_h      = ws + off; off += (size_t)BL * 64;
  float* f_trends = ws + off; off += (size_t)BL * 64;
  float* f_xn     = ws + off; off += (size_t)BL * 64;
  float* f_xz     = ws + off; off += (size_t)BL * 256;
  float* f_xc     = ws + off; off += (size_t)BL * 128;
  float* f_xbl    = ws + off; off += (size_t)BL * 48;
  float* f_delta  = ws + off; off += (size_t)BL * 128;
  float* f_ys     = ws + off; off += (size_t)BL * 128;
  float* f_yo     = ws + off; off += (size_t)BL * 64;
  float* f_cs     = ws + off; off += (size_t)256 * (LDIM + 1);
  float* f_psd    = ws + off; off += 2048;
  int*   prm      = (int*)(ws + off); off += 16;

  dim3 w32(32);
  k_zero<<<dim3((BL * 64 + 255) / 256), dim3(256), 0, stream>>>(f_trends, BL * 64);
  k_emb_conv<<<dim3(1024 * 4), w32, 0, stream>>>(x, embw, f_h);

  for (int i = 0; i < 3; ++i) {
    k_rmsnorm<<<dim3(BL / 8), dim3(256), 0, stream>>>(f_h, nw + i * 64, f_xn);
    k_wmma_gemm_nt<<<dim3(1024 * 16), w32, 0, stream>>>(
        f_xn, 64, ipw + (size_t)i * 256 * 64, 64, f_xz, 256, 64, 256, 16);
    k_dwconv_silu<<<dim3((BL * 128 + 255) / 256), dim3(256), 0, stream>>>(
        f_xz, cw + i * 128 * 4, cb + i * 128, f_xc);
    k_wmma_gemm_nt<<<dim3(1024 * 3), w32, 0, stream>>>(
        f_xc, 128, xpw + (size_t)i * 36 * 128, 128, f_xbl, 48, 128, 36, 3);
    k_delta<<<dim3(1024 * 8), w32, 0, stream>>>(
        f_xbl, dtw + i * 128 * 4, dtb + i * 128, f_delta);
    k_scan<<<dim3(256), w32, 0, stream>>>(
        f_xc, f_delta, f_xbl, f_xz, alog + i * 128 * 16, dpar + i * 128, f_ys);
    k_wmma_gemm_nt<<<dim3(1024 * 4), w32, 0, stream>>>(
        f_ys, 128, opw + (size_t)i * 64 * 128, 128, f_yo, 64, 128, 64, 4);
    k_zero<<<dim3(8), dim3(256), 0, stream>>>(f_psd, 2048);
    k_fft_psd<<<dim3(256), dim3(256), 0, stream>>>(f_yo, f_psd);
    k_argmax_period<<<dim3(1), dim3(1024), 0, stream>>>(f_psd, prm);
    k_cumsum<<<dim3(256), dim3(1024), 0, stream>>>(f_yo, f_cs);
    k_trend_update<<<dim3((BL * 64 + 255) / 256), dim3(256), 0, stream>>>(
        f_yo, f_cs, prm, f_h, f_trends);
  }
  k_final_conv<<<dim3(1024 * 8), w32, 0, stream>>>(f_trends, f_h, tw, sw, out);
}